// Median_Filter_68719477234
// MI455X (gfx1250) — compile-verified
//
#include <hip/hip_runtime.h>
#include <float.h>

// ---------------------------------------------------------------------------
// 3x3 median filter, border-clipped windows, lower-median selection.
// Shape fixed by reference: (B=64, C=3, H=224, W=224) fp32.
//
// Roofline: 38.5 MB in + 38.5 MB out = 77 MB -> ~3.3 us at 23.3 TB/s HBM
// (both tensors fit in the 192 MB L2, so graph replays run L2-resident).
// No matmul structure -> WMMA inapplicable; the win is minimizing
// instructions per byte. Each thread computes 4 horizontally adjacent
// pixels: the 3-element vertical column sorts (first 9 CEs of the verified
// columns-first 23-CE network) are shared across neighbors, leaving only
// 14 CEs per pixel. Stores are vectorized to global_store_b128.
//
// Network provenance: rows->cols->completion 25-CE sorter verified by the
// 0-1 principle over all 20 row/col-sorted 3x3 tableaux; transpose-relabeled
// (t(3i+j)=3j+i) so the shared phase is per-column sort3. Pruned CEs (5,7)
// and (5,6) only affect final ranks >= 5. Final: rank1=a3, rank2=a6,
// rank4=a4 -- exactly the lower-median ranks for corner/edge/interior
// windows under +FLT_MAX sentinel padding.
// ---------------------------------------------------------------------------

#define MF_H     224
#define MF_W     224
#define MF_TDIM  32                 // output tile: 32 x 32 pixels
#define MF_BX    8                  // threads in x, each covers 4 pixels
#define MF_BY    32                 // threads in y
#define MF_HALO  (MF_TDIM + 2)      // 34
#define MF_PITCH 36                 // pad: keeps 6-float row segs 16B aligned

// Compare-exchange: first arg <- min, second <- max.
#define CE(a, b) { float _lo = fminf(a, b); float _hi = fmaxf(a, b); (a) = _lo; (b) = _hi; }

__global__ __launch_bounds__(MF_BX * MF_BY)
void Median_Filter_68719477234_kernel(const float* __restrict__ x,
                                      float* __restrict__ out) {
    __shared__ float tile[MF_HALO * MF_PITCH];   // 34*36*4 = 4896 B

    const int z   = blockIdx.z;                  // fused (B*C) image index
    const int bh0 = blockIdx.y * MF_TDIM;
    const int bw0 = blockIdx.x * MF_TDIM;
    const int tx  = threadIdx.x;                 // 0..7
    const int ty  = threadIdx.y;                 // 0..31
    const int tid = ty * MF_BX + tx;

    const float* __restrict__ img = x + (size_t)z * (MF_H * MF_W);

    // ---- cooperative halo-tile load: 34 x 34, OOB -> +FLT_MAX sentinel ----
    // Consecutive tid -> consecutive gw -> coalesced global_load_b32.
    #pragma unroll
    for (int i = tid; i < MF_HALO * MF_HALO; i += MF_BX * MF_BY) {
        const int r  = i / MF_HALO;
        const int c  = i - r * MF_HALO;
        const int gh = bh0 - 1 + r;
        const int gw = bw0 - 1 + c;
        float v = FLT_MAX;
        if (gh >= 0 && gh < MF_H && gw >= 0 && gw < MF_W)
            v = img[gh * MF_W + gw];
        tile[r * MF_PITCH + c] = v;
    }
    __syncthreads();

    // ---- per-thread window: 4 output pixels at row h, cols w0..w0+3 ----
    const int h  = bh0 + ty;
    const int w0 = bw0 + tx * 4;
    const int cb = tx * 4;                        // halo col of (w0 - 1)

    // Three halo rows, 6 consecutive floats each (16B-aligned -> ds_load_b128+b64).
    const float* t0 = &tile[(ty + 0) * MF_PITCH + cb];
    const float* t1 = &tile[(ty + 1) * MF_PITCH + cb];
    const float* t2 = &tile[(ty + 2) * MF_PITCH + cb];

    float lo[6], mi[6], hi[6];
    #pragma unroll
    for (int c = 0; c < 6; ++c) {
        float a = t0[c], b = t1[c], d = t2[c];
        CE(a, b); CE(b, d); CE(a, b);            // shared vertical sort3
        lo[c] = a; mi[c] = b; hi[c] = d;
    }

    const int rows = (h == 0 || h == MF_H - 1) ? 2 : 3;

    float4 res;
    float* resf = reinterpret_cast<float*>(&res);
    #pragma unroll
    for (int p = 0; p < 4; ++p) {
        // Columns L=p, M=p+1, R=p+2 (each sorted): relabeled positions.
        float a0 = lo[p],     a3 = mi[p],     a6 = hi[p];
        float a1 = lo[p + 1], a4 = mi[p + 1], a7 = hi[p + 1];
        float a2 = lo[p + 2], a5 = mi[p + 2], a8 = hi[p + 2];

        // Sort the three "rows" (mins, mids, maxs across columns):
        CE(a0, a1); CE(a1, a2); CE(a0, a1);
        CE(a3, a4); CE(a4, a5); CE(a3, a4);
        CE(a6, a7); CE(a7, a8); CE(a6, a7);
        // Pruned tableau completion (relabeled; ranks <= 4 exact):
        CE(a3, a1);
        CE(a6, a2);
        CE(a4, a2);
        CE(a6, a4);
        CE(a6, a1);
        // rank1 = a3, rank2 = a6, rank4 = a4.

        const int w    = w0 + p;
        const int cols = (w == 0 || w == MF_W - 1) ? 2 : 3;
        const int idx  = (rows * cols - 1) >> 1;  // 4 interior, 2 edge, 1 corner

        float m = a4;
        if (idx == 2) m = a6;
        if (idx == 1) m = a3;
        resf[p] = m;
    }

    // 16B-aligned vector store -> global_store_b128.
    float* dst = out + (size_t)z * (MF_H * MF_W) + h * MF_W + w0;
    *reinterpret_cast<float4*>(dst) = res;
}

extern "C" void kernel_launch(void* const* d_in, const int* in_sizes, int n_in,
                              void* d_out, int out_size, void* d_ws, size_t ws_size,
                              hipStream_t stream) {
    const float* x   = (const float*)d_in[0];     // (64,3,224,224) fp32
    float*       out = (float*)d_out;             // same shape, fp32
    // d_in[1] is k_size == 3 (kernel specialized for 3x3).

    const int BC = in_sizes[0] / (MF_H * MF_W);   // 64*3 = 192 images
    dim3 block(MF_BX, MF_BY);                     // 256 threads = 8 wave32
    dim3 grid(MF_W / MF_TDIM, MF_H / MF_TDIM, BC);// (7, 7, 192)
    Median_Filter_68719477234_kernel<<<grid, block, 0, stream>>>(x, out);
}